// HeteroGAT_62680752717832
// MI455X (gfx1250) — compile-verified
//
#include <hip/hip_runtime.h>
#include <math.h>

typedef float v2f __attribute__((ext_vector_type(2)));
typedef float v8f __attribute__((ext_vector_type(8)));

#define DIN_ 128
#define NEG_SLOPE_ 0.2f
#define EPS_ 1e-16f

__device__ __forceinline__ void atomicMaxF(float* addr, float v) {
  // works for mixed-sign values when addr is initialized to -inf
  if (v >= 0.0f) atomicMax((int*)addr, __float_as_int(v));
  else           atomicMin((unsigned int*)addr, __float_as_uint(v));
}

// ---------------------------------------------------------------------------
// C[M,N] = X[M,128] @ W[128,N]   (M % 16 == 0, N in {64,128})
// f32 WMMA 16x16x4, W transposed + XOR-swizzled in LDS so each B fragment is
// a single conflict-free ds_load_b64. 8 waves/block, 16-row strip per wave.
//
// LDS layout: element W[k][col] lives at sWT[col*128 + (k ^ (4*(col&15)))].
// A fragment (16x4 f32): lane&15 = M row; lanes 0-15 hold K=k0,k0+1; lanes
// 16-31 hold K=k0+2,k0+3.  B mirrors with lane&15 = N col.  C/D: VGPR g ->
// M = g + (lane>=16)*8, N = lane&15.
// ---------------------------------------------------------------------------
__global__ __launch_bounds__(256) void gemm_xw_wmma(const float* __restrict__ X,
                                                    const float* __restrict__ W,
                                                    float* __restrict__ C,
                                                    int M, int N) {
  __shared__ float sWT[DIN_ * 128];  // exactly 64 KB
  const int tid = threadIdx.x;
  const int nshift = (N == 128) ? 7 : 6;
  for (int idx = tid; idx < DIN_ * N; idx += (int)blockDim.x) {
    int k = idx >> nshift;
    int col = idx & (N - 1);
    sWT[col * DIN_ + (k ^ ((col & 15) << 2))] = W[idx];
  }
  __syncthreads();

  const int lane = tid & 31;
  const int wave = tid >> 5;
  const int row0 = (blockIdx.x * 8 + wave) * 16;
  if (row0 >= M) return;

  const int m     = lane & 15;
  const int key   = lane & 15;         // xor key for B swizzle
  const int khalf = (lane >> 4) << 1;  // 0 or 2
  const float* xrow = X + (size_t)(row0 + m) * DIN_;
  float a0[32], a1[32];
#pragma unroll
  for (int kk = 0; kk < 32; ++kk) {
    a0[kk] = xrow[kk * 4 + khalf + 0];
    a1[kk] = xrow[kk * 4 + khalf + 1];
  }

  const int ntiles = N >> 4;
  for (int nt = 0; nt < ntiles; ++nt) {
    const int col  = nt * 16 + (lane & 15);
    const int base = col * DIN_ + khalf;  // float index of this lane's B base
    v8f c = {0.f, 0.f, 0.f, 0.f, 0.f, 0.f, 0.f, 0.f};
#pragma unroll
    for (int kk = 0; kk < 32; ++kk) {
      v2f a, b;
      a.x = a0[kk];
      a.y = a1[kk];
      // contiguous pair (k, k+1) for this col; one ds_load_b64
      b = *(const v2f*)(sWT + base + ((kk ^ key) << 2));
      c = __builtin_amdgcn_wmma_f32_16x16x4_f32(false, a, false, b, (short)0, c,
                                                false, false);
    }
    const int rbase = row0 + (lane >> 4) * 8;
#pragma unroll
    for (int g = 0; g < 8; ++g)
      C[(size_t)(rbase + g) * N + col] = c[g];
  }
}

// weff[k*H + h] = sum_f W[k, h*F + f] * a[h*F + f]    (W is [128, H*F])
__global__ void weff_kernel(const float* __restrict__ W, const float* __restrict__ a,
                            float* __restrict__ weff, int H, int F) {
  int idx = blockIdx.x * blockDim.x + threadIdx.x;
  if (idx >= DIN_ * H) return;
  int k = idx / H, h = idx % H;
  int N = H * F;
  float s = 0.f;
  for (int f = 0; f < F; ++f) s += W[k * N + h * F + f] * a[h * F + f];
  weff[idx] = s;
}

// es[i*H + h] = X[i,:] . weff[:, h]
__global__ void edot_kernel(const float* __restrict__ X, const float* __restrict__ weff,
                            float* __restrict__ es, int Nn, int H) {
  int idx = blockIdx.x * blockDim.x + threadIdx.x;
  if (idx >= Nn * H) return;
  int i = idx / H, h = idx % H;
  const float* x = X + (size_t)i * DIN_;
  float s = 0.f;
#pragma unroll 8
  for (int k = 0; k < DIN_; ++k) s += x[k] * weff[k * H + h];
  es[idx] = s;
}

__global__ void fill_kernel(float* __restrict__ p, float v, long n) {
  long i = (long)blockIdx.x * blockDim.x + threadIdx.x;
  if (i < n) p[i] = v;
}

__global__ void edge_e_kernel(const int* __restrict__ src, const int* __restrict__ dst,
                              const float* __restrict__ es, const float* __restrict__ ed,
                              float* __restrict__ eb, float* __restrict__ mb,
                              long E, int H) {
  long idx = (long)blockIdx.x * blockDim.x + threadIdx.x;
  if (idx >= E * H) return;
  long eid = idx / H;
  int h = (int)(idx % H);
  int s = src[eid], d = dst[eid];
  float v = es[(size_t)s * H + h] + ed[(size_t)d * H + h];
  v = (v > 0.f) ? v : NEG_SLOPE_ * v;
  eb[idx] = v;
  atomicMaxF(&mb[(size_t)d * H + h], v);
}

__global__ void edge_p_kernel(const int* __restrict__ dst, float* __restrict__ eb,
                              const float* __restrict__ mb, float* __restrict__ sb,
                              long E, int H) {
  long idx = (long)blockIdx.x * blockDim.x + threadIdx.x;
  if (idx >= E * H) return;
  long eid = idx / H;
  int h = (int)(idx % H);
  int d = dst[eid];
  float p = expf(eb[idx] - mb[(size_t)d * H + h]);
  eb[idx] = p;
  atomicAdd(&sb[(size_t)d * H + h], p);
}

// one thread per (edge, 4 consecutive features); F % 4 == 0 so a float4 stays
// inside one head.
__global__ void edge_msg_kernel(const int* __restrict__ src, const int* __restrict__ dst,
                                const float* __restrict__ pb, const float* __restrict__ sb,
                                const float* __restrict__ hs, float* __restrict__ out,
                                long E, int H, int F) {
  const int HF = H * F;
  const int HF4 = HF >> 2;
  long idx = (long)blockIdx.x * blockDim.x + threadIdx.x;
  if (idx >= E * (long)HF4) return;
  long eid = idx / HF4;
  int j = ((int)(idx % HF4)) << 2;
  int h = j / F;
  int s = src[eid], d = dst[eid];
  float alpha = pb[eid * H + h] / (sb[(size_t)d * H + h] + EPS_);
  const float4 hv = *(const float4*)&hs[(size_t)s * HF + j];
  float* o = &out[(size_t)d * HF + j];
  atomicAdd(o + 0, alpha * hv.x);
  atomicAdd(o + 1, alpha * hv.y);
  atomicAdd(o + 2, alpha * hv.z);
  atomicAdd(o + 3, alpha * hv.w);
}

__global__ void elu_kernel(float* __restrict__ x, long n) {
  long i = (long)blockIdx.x * blockDim.x + threadIdx.x;
  if (i >= n) return;
  float v = x[i];
  x[i] = (v > 0.f) ? v : expm1f(v);
}

// ---------------------------------------------------------------------------
// Host-side orchestration
// ---------------------------------------------------------------------------
static inline long cdiv(long a, long b) { return (a + b - 1) / b; }

static void run_relation(const float* x_src, const float* x_dst,
                         const int* src, const int* dst, long E,
                         long Ns, long Nd,
                         const float* Wsrc, const float* Wdst,
                         const float* asrc, const float* adst,
                         int H, int F,
                         float* HS, float* ES, float* ED, float* Mb, float* Sb,
                         float* Eb, float* weff_s, float* weff_d,
                         float* out, hipStream_t stream) {
  const int N = H * F;
  // hs = x_src @ Wsrc (WMMA)
  gemm_xw_wmma<<<(int)cdiv(Ns / 16, 8), 256, 0, stream>>>(x_src, Wsrc, HS, (int)Ns, N);
  // effective attention weights, then es / ed via thin dots (no hd needed)
  weff_kernel<<<(int)cdiv(DIN_ * H, 128), 128, 0, stream>>>(Wsrc, asrc, weff_s, H, F);
  weff_kernel<<<(int)cdiv(DIN_ * H, 128), 128, 0, stream>>>(Wdst, adst, weff_d, H, F);
  edot_kernel<<<(int)cdiv(Ns * H, 256), 256, 0, stream>>>(x_src, weff_s, ES, (int)Ns, H);
  edot_kernel<<<(int)cdiv(Nd * H, 256), 256, 0, stream>>>(x_dst, weff_d, ED, (int)Nd, H);
  // segment softmax over dst
  fill_kernel<<<(int)cdiv(Nd * H, 256), 256, 0, stream>>>(Mb, -INFINITY, Nd * H);
  fill_kernel<<<(int)cdiv(Nd * H, 256), 256, 0, stream>>>(Sb, 0.f, Nd * H);
  edge_e_kernel<<<(int)cdiv(E * H, 256), 256, 0, stream>>>(src, dst, ES, ED, Eb, Mb, E, H);
  edge_p_kernel<<<(int)cdiv(E * H, 256), 256, 0, stream>>>(dst, Eb, Mb, Sb, E, H);
  edge_msg_kernel<<<(int)cdiv(E * (long)(N >> 2), 256), 256, 0, stream>>>(
      src, dst, Eb, Sb, HS, out, E, H, F);
}

extern "C" void kernel_launch(void* const* d_in, const int* in_sizes, int n_in,
                              void* d_out, int out_size, void* d_ws, size_t ws_size,
                              hipStream_t stream) {
  const float* x_paper  = (const float*)d_in[0];
  const float* x_author = (const float*)d_in[1];
  const float* W1_src   = (const float*)d_in[2];  // [3][128][128]
  const float* W1_dst   = (const float*)d_in[3];
  const float* a1_src   = (const float*)d_in[4];  // [3][4][32]
  const float* a1_dst   = (const float*)d_in[5];
  const float* W2_src   = (const float*)d_in[6];  // [3][128][64]
  const float* W2_dst   = (const float*)d_in[7];
  const float* a2_src   = (const float*)d_in[8];  // [3][1][64]
  const float* a2_dst   = (const float*)d_in[9];
  const int* ci_s = (const int*)d_in[10];
  const int* ci_d = (const int*)d_in[11];
  const int* wr_s = (const int*)d_in[12];
  const int* wr_d = (const int*)d_in[13];
  const int* rv_s = (const int*)d_in[14];
  const int* rv_d = (const int*)d_in[15];

  const long NP = in_sizes[0] / DIN_;   // 100000
  const long NA = in_sizes[1] / DIN_;   // 50000
  const long Ec = in_sizes[10];
  const long Ew = in_sizes[12];
  const long Er = in_sizes[14];
  long Emax = Ec > Ew ? Ec : Ew;
  if (Er > Emax) Emax = Er;

  // workspace layout (floats)
  float* ws = (float*)d_ws;
  long off = 0;
  float* HP = ws + off; off += NP * 128;   // layer-1 paper output / layer-2 input
  float* HA = ws + off; off += NA * 128;   // layer-1 author output / layer-2 input
  float* HS = ws + off; off += NP * 128;   // per-relation hs scratch (max size)
  float* ES = ws + off; off += NP * 4;
  float* ED = ws + off; off += NP * 4;
  float* Mb = ws + off; off += NP * 4;
  float* Sb = ws + off; off += NP * 4;
  float* Eb = ws + off; off += Emax * 4;
  float* Wes = ws + off; off += DIN_ * 4;
  float* Wed = ws + off; off += DIN_ * 4;
  (void)ws_size; (void)n_in; (void)out_size;

  // ---------------- Layer 1 (H=4, F=32, HF=128) ----------------
  fill_kernel<<<(int)cdiv(NP * 128, 256), 256, 0, stream>>>(HP, 0.f, NP * 128);
  fill_kernel<<<(int)cdiv(NA * 128, 256), 256, 0, stream>>>(HA, 0.f, NA * 128);

  run_relation(x_paper, x_paper, ci_s, ci_d, Ec, NP, NP,
               W1_src + 0 * 16384, W1_dst + 0 * 16384, a1_src + 0 * 128, a1_dst + 0 * 128,
               4, 32, HS, ES, ED, Mb, Sb, Eb, Wes, Wed, HP, stream);
  run_relation(x_author, x_paper, wr_s, wr_d, Ew, NA, NP,
               W1_src + 1 * 16384, W1_dst + 1 * 16384, a1_src + 1 * 128, a1_dst + 1 * 128,
               4, 32, HS, ES, ED, Mb, Sb, Eb, Wes, Wed, HP, stream);
  run_relation(x_paper, x_author, rv_s, rv_d, Er, NP, NA,
               W1_src + 2 * 16384, W1_dst + 2 * 16384, a1_src + 2 * 128, a1_dst + 2 * 128,
               4, 32, HS, ES, ED, Mb, Sb, Eb, Wes, Wed, HA, stream);

  elu_kernel<<<(int)cdiv(NP * 128, 256), 256, 0, stream>>>(HP, NP * 128);
  elu_kernel<<<(int)cdiv(NA * 128, 256), 256, 0, stream>>>(HA, NA * 128);

  // ---------------- Layer 2 (H=1, F=64, HF=64) ----------------
  float* OP = (float*)d_out;          // [NP, 64]
  float* OA = OP + NP * 64;           // [NA, 64]
  fill_kernel<<<(int)cdiv(NP * 64, 256), 256, 0, stream>>>(OP, 0.f, NP * 64);
  fill_kernel<<<(int)cdiv(NA * 64, 256), 256, 0, stream>>>(OA, 0.f, NA * 64);

  run_relation(HP, HP, ci_s, ci_d, Ec, NP, NP,
               W2_src + 0 * 8192, W2_dst + 0 * 8192, a2_src + 0 * 64, a2_dst + 0 * 64,
               1, 64, HS, ES, ED, Mb, Sb, Eb, Wes, Wed, OP, stream);
  run_relation(HA, HP, wr_s, wr_d, Ew, NA, NP,
               W2_src + 1 * 8192, W2_dst + 1 * 8192, a2_src + 1 * 64, a2_dst + 1 * 64,
               1, 64, HS, ES, ED, Mb, Sb, Eb, Wes, Wed, OP, stream);
  run_relation(HP, HA, rv_s, rv_d, Er, NP, NA,
               W2_src + 2 * 8192, W2_dst + 2 * 8192, a2_src + 2 * 64, a2_dst + 2 * 64,
               1, 64, HS, ES, ED, Mb, Sb, Eb, Wes, Wed, OA, stream);
}